// Attention_89412629168340
// MI455X (gfx1250) — compile-verified
//
#include <hip/hip_runtime.h>

#define HDIM 1024
#define TLEN 4096
#define BATCH 16
#define MASKVAL -10000000000.0f

typedef float v2f __attribute__((ext_vector_type(2)));
typedef float v8f __attribute__((ext_vector_type(8)));

// ---------------------------------------------------------------------------
// C(16 x H) = A(16 x H) @ op(W) + bias,  W is HxH row-major.
// TRANSB=true : op(W) = W^T  (i.e. C = A @ W^T, like q = x @ Wq^T)
// TRANSB=false: op(W) = W    (i.e. C = A @ W,   like p = q @ Wk)
// One wave per 16x16 output tile, V_WMMA_F32_16X16X4_F32 along K.
// Layouts per CDNA5 ISA 7.12.2:
//   A 16x4 : lane<16 -> M=lane,  v0=K0,v1=K1 ; lane>=16 -> M=lane-16, v0=K2,v1=K3
//   B 4x16 : v0 holds rows K0(l<16)/K2(l>=16), v1 rows K1/K3, N = lane&15
//   C 16x16: VGPR j -> M = j + 8*(lane>>4), N = lane&15
// ---------------------------------------------------------------------------
template <bool TRANSB>
__global__ void gemm16xH_wmma(const float* __restrict__ A,
                              const float* __restrict__ W,
                              const float* __restrict__ bias,
                              float* __restrict__ C) {
  const int lane = threadIdx.x & 31;
  const int wid  = threadIdx.x >> 5;
  const int n0   = (blockIdx.x * (blockDim.x >> 5) + wid) * 16;
  const int m    = lane & 15;
  const int koff = (lane >> 4) << 1;  // 0 or 2
  const int n    = n0 + m;

  v8f acc = {};
#pragma unroll 8
  for (int k0 = 0; k0 < HDIM; k0 += 4) {
    v2f a = *(const v2f*)(A + m * HDIM + k0 + koff);
    v2f b;
    if (TRANSB) {
      b = *(const v2f*)(W + n * HDIM + k0 + koff);
    } else {
      b[0] = W[(k0 + koff) * HDIM + n];
      b[1] = W[(k0 + koff + 1) * HDIM + n];
    }
    acc = __builtin_amdgcn_wmma_f32_16x16x4_f32(false, a, false, b,
                                                (short)0, acc, false, false);
  }

  const float bn = bias ? bias[n] : 0.0f;
  const int mb = (lane >> 4) << 3;  // 0 or 8
#pragma unroll
  for (int j = 0; j < 8; ++j) {
    C[(mb + j) * HDIM + n] = acc[j] + bn;
  }
}

// qin[b][h] = inputs[(lengths[b]-1), b, h]
__global__ void gather_kernel(const float* __restrict__ x,
                              const int* __restrict__ lengths,
                              float* __restrict__ qin) {
  int idx = blockIdx.x * 256 + threadIdx.x;      // 0 .. B*H-1
  int b = idx >> 10;
  int h = idx & (HDIM - 1);
  int t = lengths[b] - 1;
  qin[idx] = x[((size_t)t * BATCH + b) * HDIM + h];
}

// out[b] = q_b . bk
__global__ void dotbk_kernel(const float* __restrict__ q,
                             const float* __restrict__ bk,
                             float* __restrict__ out) {
  __shared__ float red[256];
  int b = blockIdx.x, tid = threadIdx.x;
  float s = 0.f;
  for (int h = tid; h < HDIM; h += 256) s += q[b * HDIM + h] * bk[h];
  red[tid] = s;
  __syncthreads();
  for (int o = 128; o > 0; o >>= 1) {
    if (tid < o) red[tid] += red[tid + o];
    __syncthreads();
  }
  if (tid == 0) out[b] = red[0];
}

// scores[b][t] = mask( (p_b . x_bt + qdotbk[b]) / 32 ).  One wave per (b,t).
__global__ void scores_kernel(const float* __restrict__ x,
                              const float* __restrict__ p,
                              const float* __restrict__ qdotbk,
                              const int* __restrict__ lengths,
                              float* __restrict__ scores) {
  int w = blockIdx.x * (blockDim.x >> 5) + (threadIdx.x >> 5);
  int lane = threadIdx.x & 31;
  int t = w / BATCH;
  int b = w - t * BATCH;
  const float* xr = x + ((size_t)t * BATCH + b) * HDIM;
  const float* pr = p + b * HDIM;
  float s = 0.f;
#pragma unroll
  for (int j = 0; j < HDIM; j += 128) {
    const float4 xv = *(const float4*)(xr + j + lane * 4);
    const float4 pv = *(const float4*)(pr + j + lane * 4);
    s += xv.x * pv.x + xv.y * pv.y + xv.z * pv.z + xv.w * pv.w;
  }
#pragma unroll
  for (int off = 16; off > 0; off >>= 1) s += __shfl_xor(s, off, 32);
  if (lane == 0) {
    float val = (t < lengths[b]) ? (s + qdotbk[b]) * 0.03125f : MASKVAL;
    scores[(size_t)b * TLEN + t] = val;
  }
}

// Per-batch masked softmax over T; writes weights to workspace AND d_out.
__global__ void softmax_kernel(const float* __restrict__ scores,
                               float* __restrict__ weights,
                               float* __restrict__ attn_out) {
  __shared__ float red[256];
  int b = blockIdx.x, tid = threadIdx.x;
  const float* s = scores + (size_t)b * TLEN;

  float mx = -3.4e38f;
  for (int t = tid; t < TLEN; t += 256) mx = fmaxf(mx, s[t]);
  red[tid] = mx;
  __syncthreads();
  for (int o = 128; o > 0; o >>= 1) {
    if (tid < o) red[tid] = fmaxf(red[tid], red[tid + o]);
    __syncthreads();
  }
  mx = red[0];
  __syncthreads();

  float sum = 0.f;
  for (int t = tid; t < TLEN; t += 256) sum += __expf(s[t] - mx);
  red[tid] = sum;
  __syncthreads();
  for (int o = 128; o > 0; o >>= 1) {
    if (tid < o) red[tid] += red[tid + o];
    __syncthreads();
  }
  float inv = 1.0f / red[0];

  for (int t = tid; t < TLEN; t += 256) {
    float wv = __expf(s[t] - mx) * inv;
    weights[(size_t)b * TLEN + t] = wv;
    attn_out[(size_t)b * TLEN + t] = wv;
  }
}

// xwp[tc][b][h] = sum over the tc-th T-chunk of w[b,t] * x[t,b,h]
__global__ void xw_partial_kernel(const float* __restrict__ x,
                                  const float* __restrict__ w,
                                  float* __restrict__ xwp) {
  int b = blockIdx.x;
  int h = blockIdx.y * 256 + threadIdx.x;
  int tc = blockIdx.z;
  const int tlen = TLEN / 8;
  const float* wr = w + (size_t)b * TLEN + tc * tlen;
  const float* xr = x + ((size_t)(tc * tlen) * BATCH + b) * HDIM + h;
  float acc = 0.f;
  for (int t = 0; t < tlen; ++t) {
    acc += wr[t] * xr[(size_t)t * BATCH * HDIM];
  }
  xwp[((size_t)tc * BATCH + b) * HDIM + h] = acc;
}

__global__ void xw_reduce_kernel(const float* __restrict__ xwp,
                                 float* __restrict__ xw) {
  int idx = blockIdx.x * 256 + threadIdx.x;  // 0 .. B*H-1
  float s = 0.f;
#pragma unroll
  for (int c = 0; c < 8; ++c) s += xwp[(size_t)c * (BATCH * HDIM) + idx];
  xw[idx] = s;
}

extern "C" void kernel_launch(void* const* d_in, const int* in_sizes, int n_in,
                              void* d_out, int out_size, void* d_ws, size_t ws_size,
                              hipStream_t stream) {
  const float* inputs = (const float*)d_in[0];
  const int* lengths  = (const int*)d_in[1];
  const float* Wq = (const float*)d_in[2];
  const float* bq = (const float*)d_in[3];
  const float* Wk = (const float*)d_in[4];
  const float* bk = (const float*)d_in[5];
  const float* Wv = (const float*)d_in[6];
  const float* bv = (const float*)d_in[7];
  const float* We = (const float*)d_in[8];
  const float* be = (const float*)d_in[9];
  float* out = (float*)d_out;
  float* ws  = (float*)d_ws;

  // workspace layout (floats), all offsets 64B-aligned
  float* qin     = ws;                 // 16384
  float* q       = ws + 16384;         // 16384
  float* p       = ws + 32768;         // 16384
  float* qdotbk  = ws + 49152;         // 16
  float* scores  = ws + 49168;         // 65536
  float* weights = ws + 114704;        // 65536
  float* xwp     = ws + 180240;        // 131072
  float* xw      = ws + 311312;        // 16384
  float* vw      = ws + 327696;        // 16384

  // 1. gather last-valid-token rows
  gather_kernel<<<(BATCH * HDIM) / 256, 256, 0, stream>>>(inputs, lengths, qin);
  // 2. q = qin @ Wq^T + bq            (WMMA fp32)
  gemm16xH_wmma<true><<<16, 128, 0, stream>>>(qin, Wq, bq, q);
  // 3. p = q @ Wk (no bias)           (WMMA fp32)
  gemm16xH_wmma<false><<<16, 128, 0, stream>>>(q, Wk, nullptr, p);
  // 4. qdotbk[b] = q_b . bk
  dotbk_kernel<<<BATCH, 256, 0, stream>>>(q, bk, qdotbk);
  // 5. masked scores: one streaming pass over inputs (268 MB)
  scores_kernel<<<(BATCH * TLEN) / 8, 256, 0, stream>>>(inputs, p, qdotbk,
                                                        lengths, scores);
  // 6. softmax -> attention weights (also written to d_out tail)
  softmax_kernel<<<BATCH, 256, 0, stream>>>(scores, weights,
                                            out + BATCH * HDIM);
  // 7. xw[b,h] = sum_t w[b,t] x[t,b,h]: second streaming pass (268 MB)
  xw_partial_kernel<<<dim3(BATCH, HDIM / 256, 8), 256, 0, stream>>>(inputs,
                                                                    weights,
                                                                    xwp);
  xw_reduce_kernel<<<(BATCH * HDIM) / 256, 256, 0, stream>>>(xwp, xw);
  // 8. vw  = xw @ Wv^T + bv           (WMMA fp32)
  gemm16xH_wmma<true><<<16, 128, 0, stream>>>(xw, Wv, bv, vw);
  // 9. emb = vw @ We^T + be -> d_out  (WMMA fp32)
  gemm16xH_wmma<true><<<16, 128, 0, stream>>>(vw, We, be, out);
}